// CBF_19000935317924
// MI455X (gfx1250) — compile-verified
//
#include <hip/hip_runtime.h>
#include <hip/hip_bf16.h>
#include <math.h>

typedef float v2f __attribute__((ext_vector_type(2)));
typedef float v8f __attribute__((ext_vector_type(8)));

// ---- problem constants ----
#define N_STATE 12
#define HID     128
#define B_TILE  16           // batch rows per tile
#define M_TILES 13           // 1 V tile + 12 JV tiles
#define WSTR    132          // padded LDS row stride (floats) -> conflict-free b64
#define TILES_PER_WG 2       // batch tiles per workgroup (amortize weight staging)

// ---- LDS layout (float offsets) ----
#define OFF_W     0                       // 3 * 128*132 = 50688
#define OFF_A     50688                   // 208*132 = 27456 -> 78144
#define OFF_W0    78144                   // 128*12 = 1536  -> 79680
#define OFF_BIAS  79680                   // 4*128 = 512    -> 80192
#define OFF_WOUT  80192                   // 128            -> 80320
#define OFF_BOUT  80320                   // 1 (+3 pad)     -> 80324
#define OFF_XN    80324                   // 16*12 = 192    -> 80516
#define OFF_INVR  80516                   // 12             -> 80528
#define SMEM_FLOATS 80528                 // 322,112 bytes  (<= 320KB WGP LDS)

#if defined(__has_builtin)
#if __has_builtin(__builtin_amdgcn_tensor_load_to_lds)
#define USE_TDM 1
#endif
#endif

#ifdef USE_TDM
typedef unsigned int v4u __attribute__((ext_vector_type(4)));
typedef int          v8i __attribute__((ext_vector_type(8)));
typedef int          v4i __attribute__((ext_vector_type(4)));

// DMA one 128x128 f32 row-major matrix from global into LDS, padding the
// destination by 4 DWORDs every 128 DWORDs (-> row stride 132 floats).
// D# packing per CDNA5 ISA.
__device__ __forceinline__ void tdm_load_weight(const float* gsrc, const float* ldst)
{
    unsigned long long ga  = (unsigned long long)(const void*)gsrc;
    unsigned int       lds = (unsigned int)(unsigned long long)(const void*)ldst; // addr[31:0] = LDS byte offset

    v4u g0;
    g0[0] = 1u;                                   // count=1 (valid user descriptor)
    g0[1] = lds;                                  // lds_addr
    g0[2] = (unsigned int)ga;                     // global_addr[31:0]
    g0[3] = (unsigned int)((ga >> 32) & 0x1FFFFFFu) | (2u << 30); // global_addr[56:32] | type=2

    v8i g1;
    g1[0] = (2 << 16)      // data_size = 4 bytes
          | (1 << 20)      // pad_enable
          | (6 << 22)      // pad_interval: 128 DWORDs
          | (3 << 25);     // pad_amount:   4 DWORDs
    g1[1] = (HID << 16);   // atomic_barrier_addr=0 | tensor_dim0[15:0]=128
    g1[2] = (HID << 16);   // tensor_dim0[31:16]=0  | tensor_dim1[15:0]=128
    g1[3] = (HID << 16);   // tensor_dim1[31:16]=0  | tile_dim0=128
    g1[4] = HID;           // tile_dim1=128 | tile_dim2=0
    g1[5] = HID;           // tensor_dim0_stride[31:0]=128
    g1[6] = 0;             // stride hi | tensor_dim1_stride lo
    g1[7] = 0;

    v4i zz = {0, 0, 0, 0};
#if __clang_major__ >= 23
    v8i zz8 = {0, 0, 0, 0, 0, 0, 0, 0};
    __builtin_amdgcn_tensor_load_to_lds(g0, g1, zz, zz, zz8, 0);
#else
    __builtin_amdgcn_tensor_load_to_lds(g0, g1, zz, zz, 0);
#endif
}
#endif // USE_TDM

extern "C" __global__ void __launch_bounds__(256)
cbf_mlp_jac_kernel(const float* __restrict__ state,
                   const float* __restrict__ safem,
                   const float* __restrict__ safel,
                   const float* __restrict__ W0g,  const float* __restrict__ b0g,
                   const float* __restrict__ W1g,  const float* __restrict__ b1g,
                   const float* __restrict__ W2g,  const float* __restrict__ b2g,
                   const float* __restrict__ W3g,  const float* __restrict__ b3g,
                   const float* __restrict__ Woutg,const float* __restrict__ boutg,
                   float* __restrict__ out)
{
    extern __shared__ float smem[];
    const int tid = threadIdx.x;

    // ---------------- stage weights (TDM async) + small tables ----------------
#ifdef USE_TDM
    if (tid < 32) {
        // one wave issues the three async weight DMAs; overlaps with layer-0 work
        tdm_load_weight(W1g, smem + OFF_W + 0 * (HID * WSTR));
        tdm_load_weight(W2g, smem + OFF_W + 1 * (HID * WSTR));
        tdm_load_weight(W3g, smem + OFF_W + 2 * (HID * WSTR));
    }
#else
    {
        const float* Ws[3] = { W1g, W2g, W3g };
        #pragma unroll
        for (int L = 0; L < 3; ++L) {
            const float* Wg = Ws[L];
            float* dst = smem + OFF_W + L * (HID * WSTR);
            for (int i = tid; i < HID * HID; i += 256) {
                int r = i >> 7, c = i & 127;
                dst[r * WSTR + c] = Wg[i];
            }
        }
    }
#endif
    {
        for (int i = tid; i < HID * N_STATE; i += 256) smem[OFF_W0 + i] = W0g[i];
        for (int i = tid; i < 4 * HID; i += 256) {
            int L = i >> 7, c = i & 127;
            const float* bg = (L == 0) ? b0g : (L == 1) ? b1g : (L == 2) ? b2g : b3g;
            smem[OFF_BIAS + i] = bg[c];
        }
        if (tid < HID) smem[OFF_WOUT + tid] = Woutg[tid];
        if (tid == 0)  smem[OFF_BOUT] = boutg[0];
        if (tid < N_STATE) {
            float m = safem[tid], l = safel[tid];
            smem[OFF_INVR + tid] = 1.0f / ((m - l) * 0.5f);
        }
    }
    __syncthreads();   // W0 / biases / invR visible

    const int wave = tid >> 5;
    const int lane = tid & 31;
    const int hi   = lane >> 4;       // 0: lanes 0-15, 1: lanes 16-31
    const int nloc = lane & 15;
    const int c    = wave * 16 + nloc;   // output column this lane owns in C/D tiles

    v8f   acc[M_TILES];
    float tv[8];

    #pragma clang loop unroll(disable)
    for (int tile = 0; tile < TILES_PER_WG; ++tile) {
        const int bbase = (blockIdx.x * TILES_PER_WG + tile) * B_TILE;

        // stage x_norm for this batch tile
        for (int i = tid; i < B_TILE * N_STATE; i += 256) {
            int r = i / N_STATE, n = i % N_STATE;
            float m = safem[n], l = safel[n];
            float xc = (m + l) * 0.5f, xr = (m - l) * 0.5f;
            smem[OFF_XN + i] = (state[(bbase + r) * N_STATE + n] - xc) / xr;
        }
        __syncthreads();

        // ------------ layer 0: K=12, VALU (JV0 = W0[:,n]/x_range[n]) ------------
        #pragma unroll
        for (int v = 0; v < 8; ++v) {
            float s = smem[OFF_BIAS + 0 * HID + c];
            const int r = v + 8 * hi;
            #pragma unroll
            for (int n = 0; n < N_STATE; ++n)
                s += smem[OFF_XN + r * N_STATE + n] * smem[OFF_W0 + c * N_STATE + n];
            acc[0][v] = s;
        }
        #pragma unroll
        for (int t = 1; t < M_TILES; ++t) {
            float val = smem[OFF_W0 + c * N_STATE + (t - 1)] * smem[OFF_INVR + (t - 1)];
            #pragma unroll
            for (int v = 0; v < 8; ++v) acc[t][v] = val;
        }
        // input_activation: tanh on V, scale JV by 1 - tanh^2 (register-local)
        #pragma unroll
        for (int v = 0; v < 8; ++v) {
            float y = tanhf(acc[0][v]);
            acc[0][v] = y;
            tv[v] = 1.0f - y * y;
        }
        #pragma unroll
        for (int t = 1; t < M_TILES; ++t)
            #pragma unroll
            for (int v = 0; v < 8; ++v) acc[t][v] *= tv[v];
        #pragma unroll
        for (int t = 0; t < M_TILES; ++t)
            #pragma unroll
            for (int v = 0; v < 8; ++v)
                smem[OFF_A + (16 * t + v + 8 * hi) * WSTR + c] = acc[t][v];

#ifdef USE_TDM
        if (tile == 0 && tid < 32)
            __builtin_amdgcn_s_wait_tensorcnt(0);   // weights landed in LDS
#endif
        __syncthreads();

        // ------------ layers 1..3: full 128x128 WMMA (fp32, 16x16x4) ------------
        #pragma clang loop unroll(disable)
        for (int L = 1; L <= 3; ++L) {
            const float* Wl   = smem + OFF_W + (L - 1) * (HID * WSTR);
            const float* Brow = Wl + c * WSTR;                 // B[k][n=c] = W[c][k]
            const float* Arow = smem + OFF_A + nloc * WSTR;    // row m=nloc within tile

            const float bias_c = smem[OFF_BIAS + L * HID + c];
            #pragma unroll
            for (int v = 0; v < 8; ++v) acc[0][v] = bias_c;    // bias only on V rows
            #pragma unroll
            for (int t = 1; t < M_TILES; ++t)
                #pragma unroll
                for (int v = 0; v < 8; ++v) acc[t][v] = 0.0f;

            for (int ks = 0; ks < 32; ++ks) {
                const int k0 = ks * 4 + 2 * hi;
                const v2f bfrag = *reinterpret_cast<const v2f*>(Brow + k0);
                #pragma unroll
                for (int t = 0; t < M_TILES; ++t) {
                    const v2f afrag =
                        *reinterpret_cast<const v2f*>(Arow + t * (16 * WSTR) + k0);
                    acc[t] = __builtin_amdgcn_wmma_f32_16x16x4_f32(
                        false, afrag, false, bfrag, (short)0, acc[t], false, false);
                }
            }
            __syncthreads();   // all waves finished reading A

            if (L < 3) {       // layer_0/1 activation; layer_2_linear has no tanh
                #pragma unroll
                for (int v = 0; v < 8; ++v) {
                    float y = tanhf(acc[0][v]);
                    acc[0][v] = y;
                    tv[v] = 1.0f - y * y;
                }
                #pragma unroll
                for (int t = 1; t < M_TILES; ++t)
                    #pragma unroll
                    for (int v = 0; v < 8; ++v) acc[t][v] *= tv[v];
            }
            #pragma unroll
            for (int t = 0; t < M_TILES; ++t)
                #pragma unroll
                for (int v = 0; v < 8; ++v)
                    smem[OFF_A + (16 * t + v + 8 * hi) * WSTR + c] = acc[t][v];
            __syncthreads();
        }

        // -------- output layer: Wout (1x128) + concat [V, JV[:,0,:]] --------
        if (tid < M_TILES * B_TILE) {
            const int row = tid;
            const int t   = row >> 4;        // 0 = V, 1..12 = tangent dir (t-1)
            const int r   = row & 15;
            float s = (t == 0) ? smem[OFF_BOUT] : 0.0f;
            const float* arow = smem + OFF_A + row * WSTR;
            #pragma unroll 8
            for (int h = 0; h < HID; ++h) s += arow[h] * smem[OFF_WOUT + h];
            const int b = bbase + r;
            out[b * 13 + t] = s;             // col 0 = V, col t (=1+d) = JV dir d
        }
        __syncthreads();   // protect A / x_norm before next batch tile
    }
}

extern "C" void kernel_launch(void* const* d_in, const int* in_sizes, int n_in,
                              void* d_out, int out_size, void* d_ws, size_t ws_size,
                              hipStream_t stream) {
    const float* state = (const float*)d_in[0];
    const float* safem = (const float*)d_in[1];
    const float* safel = (const float*)d_in[2];
    const float* W0    = (const float*)d_in[3];
    const float* b0    = (const float*)d_in[4];
    const float* W1    = (const float*)d_in[5];
    const float* b1    = (const float*)d_in[6];
    const float* W2    = (const float*)d_in[7];
    const float* b2    = (const float*)d_in[8];
    const float* W3    = (const float*)d_in[9];
    const float* b3    = (const float*)d_in[10];
    const float* Wout  = (const float*)d_in[11];
    const float* bout  = (const float*)d_in[12];
    float* out = (float*)d_out;

    const int bs   = in_sizes[0] / N_STATE;
    const int grid = bs / (B_TILE * TILES_PER_WG);
    const size_t shmem = (size_t)SMEM_FLOATS * sizeof(float);

    (void)hipFuncSetAttribute((const void*)cbf_mlp_jac_kernel,
                              hipFuncAttributeMaxDynamicSharedMemorySize,
                              (int)shmem);

    cbf_mlp_jac_kernel<<<grid, 256, shmem, stream>>>(
        state, safem, safel, W0, b0, W1, b1, W2, b2, W3, b3, Wout, bout, out);
}